// BiasVectorsBlock_68109591380070
// MI455X (gfx1250) — compile-verified
//
#include <hip/hip_runtime.h>
#include <hip/hip_bf16.h>

#include <stdint.h>

typedef __attribute__((ext_vector_type(2))) float v2f;
typedef __attribute__((ext_vector_type(8))) float v8f;

#define BB 32
#define TT 2048
#define DD 512
#define KTOT (BB * TT)      // 65536 flattened (b,t) rows
#define NS 32               // number of MVN draws
#define KSPLIT 4
#define LDS_STRIDE 68       // 64 + 4: rows 272B (16B-aligned for b128), 4-bank shift/row

// ---------------- per-batch time means: meanB[b][d] = mean_t x[b,t,d] ----------
__global__ __launch_bounds__(512)
void meanb_kernel(const float* __restrict__ x, float* __restrict__ meanB) {
  const int b = blockIdx.x;
  const int d = threadIdx.x;
  const float* xb = x + (size_t)b * TT * DD + d;
  float s = 0.0f;
  for (int t = 0; t < TT; ++t) s += xb[(size_t)t * DD];
  meanB[b * DD + d] = s * (1.0f / TT);
}

// ---------------- global mean over batches --------------------------------------
__global__ __launch_bounds__(512)
void mean_kernel(const float* __restrict__ meanB, float* __restrict__ mean) {
  const int d = threadIdx.x;
  float s = 0.0f;
  for (int b = 0; b < BB; ++b) s += meanB[b * DD + d];
  mean[d] = s * (1.0f / BB);
}

// ---------------- Gram matrix via V_WMMA_F32_16X16X4_F32 ------------------------
// covP[kz][d][e] = sum_{k in chunk kz} xc[k][d] * xc[k][e]
//
// 64x64 output tile per block, 8 waves, each wave owns two 16x16 subtiles in the
// same row band (shared A fragment). Raw x tiles are staged global->LDS with
// async b128 copies (ASYNCcnt, double-buffered); the per-batch mean is folded
// into the fragment build as a per-lane constant subtract (column-only
// dependence), which is exactly the centered product (x_kd-m_d)(x_ke-m_e).
// A 64-row k-tile never crosses a batch boundary (64 | 2048).
__global__ __launch_bounds__(256)
void gram_kernel(const float* __restrict__ x, const float* __restrict__ meanB,
                 float* __restrict__ covP) {
  __shared__ float lA[2][64 * LDS_STRIDE];
  __shared__ float lB[2][64 * LDS_STRIDE];

  const int d0 = blockIdx.y * 64;   // output row block
  const int e0 = blockIdx.x * 64;   // output col block
  const int kz = blockIdx.z;        // K split index

  const int tid  = threadIdx.x;
  const int lane = tid & 31;
  const int w    = tid >> 5;        // wave 0..7
  const int half = lane >> 4;       // 0: lanes 0-15, 1: lanes 16-31
  const int m    = lane & 15;

  const int subR  = (w & 3) * 16;          // subtile row offset within 64
  const int subC0 = ((w >> 2) * 2) * 16;   // first subtile col offset
  const int subC1 = subC0 + 16;

  v8f acc0 = {};
  v8f acc1 = {};

  // cooperative loader mapping: 256 threads, 16 rows x 16 float4 per pass
  const int lr = tid >> 4;         // row within 16-row pass
  const int lc = (tid & 15) * 4;   // starting column (float4)

  const int kBase  = kz * (KTOT / KSPLIT);
  const int nTiles = (KTOT / KSPLIT) / 64;   // 256 k-tiles per block

  // issue 8 async b128 copies (4 rows x {A,B}) for one 64-row k-tile
  auto issue_tile = [&](int k0, int buf) {
    for (int rr = 0; rr < 64; rr += 16) {
      const int row = rr + lr;
      const float* gA = x + (size_t)(k0 + row) * DD + d0 + lc;
      const float* gB = x + (size_t)(k0 + row) * DD + e0 + lc;
      const unsigned la = (unsigned)(uintptr_t)(&lA[buf][row * LDS_STRIDE + lc]);
      const unsigned lb = (unsigned)(uintptr_t)(&lB[buf][row * LDS_STRIDE + lc]);
      asm volatile("global_load_async_to_lds_b128 %0, %1, off"
                   :: "v"(la), "v"(gA) : "memory");
      asm volatile("global_load_async_to_lds_b128 %0, %1, off"
                   :: "v"(lb), "v"(gB) : "memory");
    }
  };

  issue_tile(kBase, 0);   // prologue: stage first tile

  for (int t = 0; t < nTiles; ++t) {
    const int k0 = kBase + t * 64;
    const int b  = k0 >> 11;   // batch index, constant over the 64-row tile

    // per-lane column means (same address for 32 consecutive tiles -> L0 hit)
    const float mA  = meanB[b * DD + d0 + subR  + m];
    const float mB0 = meanB[b * DD + e0 + subC0 + m];
    const float mB1 = meanB[b * DD + e0 + subC1 + m];

    const int buf = t & 1;
    if (t + 1 < nTiles) {
      issue_tile(kBase + (t + 1) * 64, buf ^ 1);        // prefetch next tile
      asm volatile("s_wait_asynccnt 0x8" ::: "memory"); // tile t complete
    } else {
      asm volatile("s_wait_asynccnt 0x0" ::: "memory");
    }
    __syncthreads();   // all waves' async copies for tile t visible

    const float* A  = &lA[buf][0];
    const float* Bt = &lB[buf][0];
    for (int kk = 0; kk < 64; kk += 4) {
      // A (16x4, ISA layout): lanes 0-15 -> K=kk+0 (v0), kk+1 (v1);
      //                       lanes 16-31 -> K=kk+2 (v0), kk+3 (v1); M = lane%16
      const int ka = kk + half * 2;
      v2f a;
      a.x = A[(ka + 0) * LDS_STRIDE + subR + m] - mA;
      a.y = A[(ka + 1) * LDS_STRIDE + subR + m] - mA;

      // B (4x16): rows striped across lanes within a VGPR:
      //   v0: K=kk+0 (lanes 0-15) / kk+1 (lanes 16-31)
      //   v1: K=kk+2 (lanes 0-15) / kk+3 (lanes 16-31); N = lane%16
      const int kb = kk + half;
      v2f b0, b1;
      b0.x = Bt[(kb + 0) * LDS_STRIDE + subC0 + m] - mB0;
      b0.y = Bt[(kb + 2) * LDS_STRIDE + subC0 + m] - mB0;
      b1.x = Bt[(kb + 0) * LDS_STRIDE + subC1 + m] - mB1;
      b1.y = Bt[(kb + 2) * LDS_STRIDE + subC1 + m] - mB1;

      acc0 = __builtin_amdgcn_wmma_f32_16x16x4_f32(
          false, a, false, b0, (short)0, acc0, false, false);
      acc1 = __builtin_amdgcn_wmma_f32_16x16x4_f32(
          false, a, false, b1, (short)0, acc1, false, false);
    }
    __syncthreads();   // done with buf before it is overwritten at step t+1
  }

  // C/D layout: VGPR r -> M = r (lanes 0-15) or r+8 (lanes 16-31), N = lane%16
  float* outp = covP + (size_t)kz * DD * DD;
  for (int r = 0; r < 8; ++r) {
    const int row = d0 + subR + r + half * 8;
    outp[(size_t)row * DD + e0 + subC0 + m] = acc0[r];
    outp[(size_t)row * DD + e0 + subC1 + m] = acc1[r];
  }
}

// ---------------- deterministic K-split reduction + scaling ---------------------
__global__ __launch_bounds__(256)
void covreduce_kernel(const float* __restrict__ covP, float* __restrict__ cov) {
  const int i = blockIdx.x * blockDim.x + threadIdx.x;  // 0 .. DD*DD-1
  float s = 0.0f;
  for (int kz = 0; kz < KSPLIT; ++kz) s += covP[(size_t)kz * DD * DD + i];
  cov[i] = s * (1.0f / ((float)(TT - 1) * (float)BB));
}

// ---------------- single-workgroup right-looking Cholesky (in place) ------------
__global__ __launch_bounds__(512)
void cholesky_kernel(float* __restrict__ A) {
  const int tid = threadIdx.x;
  for (int k = 0; k < DD; ++k) {
    if (tid == 0) A[k * DD + k] = sqrtf(A[k * DD + k]);
    __syncthreads();
    const float dk = A[k * DD + k];
    for (int i = k + 1 + tid; i < DD; i += 512) A[(size_t)i * DD + k] /= dk;
    __syncthreads();
    for (int i = k + 1 + tid; i < DD; i += 512) {
      const float lik = A[(size_t)i * DD + k];
      float* Ai = A + (size_t)i * DD;
      for (int j = k + 1; j <= i; ++j) Ai[j] -= lik * A[(size_t)j * DD + k];
    }
    __syncthreads();
  }
}

// ---------------- out[s][d] = mean[d] + sum_{e<=d} z[s][e] * L[d][e] ------------
__global__ __launch_bounds__(256)
void draw_kernel(const float* __restrict__ z, const float* __restrict__ L,
                 const float* __restrict__ mean, float* __restrict__ out) {
  const int idx = blockIdx.x * blockDim.x + threadIdx.x;
  if (idx >= NS * DD) return;
  const int s = idx >> 9;          // draw index
  const int d = idx & (DD - 1);    // feature index
  const float* zs = z + s * DD;
  const float* Ld = L + (size_t)d * DD;
  float acc = 0.0f;
  for (int e = 0; e <= d; ++e) acc += zs[e] * Ld[e];  // L is lower-triangular
  out[idx] = mean[d] + acc;
}

extern "C" void kernel_launch(void* const* d_in, const int* in_sizes, int n_in,
                              void* d_out, int out_size, void* d_ws, size_t ws_size,
                              hipStream_t stream) {
  (void)in_sizes; (void)n_in; (void)out_size; (void)ws_size;
  const float* x = (const float*)d_in[0];   // [32, 2048, 512]
  const float* z = (const float*)d_in[1];   // [32, 512]
  float* out = (float*)d_out;               // [32, 512]

  float* ws    = (float*)d_ws;
  float* meanB = ws;                                  // 32*512
  float* mean  = ws + BB * DD;                        // 512
  float* covP  = mean + DD;                           // KSPLIT * 512*512
  float* cov   = covP + (size_t)KSPLIT * DD * DD;     // 512*512

  meanb_kernel<<<BB, DD, 0, stream>>>(x, meanB);
  mean_kernel<<<1, DD, 0, stream>>>(meanB, mean);

  dim3 g(DD / 64, DD / 64, KSPLIT);                   // (8, 8, 4)
  gram_kernel<<<g, 256, 0, stream>>>(x, meanB, covP);

  covreduce_kernel<<<(DD * DD) / 256, 256, 0, stream>>>(covP, cov);
  cholesky_kernel<<<1, 512, 0, stream>>>(cov);
  draw_kernel<<<(NS * DD + 255) / 256, 256, 0, stream>>>(z, cov, mean, out);
}